// CPEmbedding_84533546320228
// MI455X (gfx1250) — compile-verified
//
#include <hip/hip_runtime.h>

typedef float v2f __attribute__((ext_vector_type(2)));
typedef float v8f __attribute__((ext_vector_type(8)));

#define RANK           32
#define EMB            512
#define TOK_PER_BLOCK  128     // 8 waves x 16 tokens
#define COLS_PER_BLOCK 128     // 8 col-tiles of 16
#define LDS_STRIDE     35      // odd, conflict-free for frag reads
#define PADDING_IDX    0

__global__ __launch_bounds__(256)
void cp_embed_wmma_f32(const int*   __restrict__ x,
                       const float* __restrict__ U0, const float* __restrict__ U1,
                       const float* __restrict__ U2, const float* __restrict__ U3,
                       const float* __restrict__ U4, const float* __restrict__ U5,
                       float*       __restrict__ out)
{
    __shared__ float Esub[COLS_PER_BLOCK * LDS_STRIDE];  // E[col][r] slice
    __shared__ float Wt  [TOK_PER_BLOCK * LDS_STRIDE];   // w[token][r]

    const int tid     = threadIdx.x;
    const int wave    = tid >> 5;
    const int lane    = tid & 31;
    const int tokbase = blockIdx.x * TOK_PER_BLOCK;
    const int colbase = blockIdx.y * COLS_PER_BLOCK;

    // ---- Stage 1: build the 128-column slice of E = U3 (x) U4 (x) U5 -------
    // E[col][r] = U3[d][r]*U4[e][r]*U5[f][r], col = d*64 + e*8 + f
    for (int i = tid; i < COLS_PER_BLOCK * RANK; i += 256) {
        const int cl  = i >> 5;          // local column
        const int r   = i & 31;
        const int col = colbase + cl;
        const int d   = col >> 6;
        const int e   = (col >> 3) & 7;
        const int f   = col & 7;
        Esub[cl * LDS_STRIDE + r] =
            U3[d * RANK + r] * U4[e * RANK + r] * U5[f * RANK + r];
    }

    // ---- Stage 2: per-wave token weights w[m][r] = U0[a][r]*U1[b][r]*U2[c][r]
    {
        const int m     = lane & 15;
        const int rbase = (lane >> 4) * 16;     // lanes 0-15: r 0..15, 16-31: r 16..31
        const int trow  = wave * 16 + m;
        const int v     = x[tokbase + trow];
        const int a     = v / 1000;             // VOC_QUANT = [50,40,25]
        const int b     = (v / 25) % 40;
        const int c     = v % 25;
        const float s   = (v == PADDING_IDX) ? 0.0f : 1.0f;  // branchless pad-zero
        #pragma unroll
        for (int r2 = 0; r2 < 16; ++r2) {
            const int r = rbase + r2;
            Wt[trow * LDS_STRIDE + r] =
                s * U0[a * RANK + r] * U1[b * RANK + r] * U2[c * RANK + r];
        }
    }
    __syncthreads();

    // ---- Stage 3: (16 tok x 32) x (32 x 128 col) via V_WMMA_F32_16X16X4_F32
    // A 16x4 f32 layout: lane = M&15, half selects K {0,1} vs {2,3}
    const int m     = lane & 15;
    const int koff  = (lane >> 4) << 1;          // 0 or 2
    const int wrow  = wave * 16 + m;
    const int mbase = (lane >> 4) * 8;           // D: VGPR v -> row v + mbase

    v2f afrag[8];
    #pragma unroll
    for (int k = 0; k < 8; ++k) {
        const float* p = &Wt[wrow * LDS_STRIDE + k * 4 + koff];
        afrag[k].x = p[0];
        afrag[k].y = p[1];
    }

    #pragma unroll
    for (int nt = 0; nt < 8; ++nt) {             // 8 col-tiles of 16
        v8f acc = {};
        #pragma unroll
        for (int k = 0; k < 8; ++k) {            // K = 32 in chunks of 4
            v2f bfrag;                           // B 4x16: lane = N, mirror of A
            const float* p = &Esub[(nt * 16 + m) * LDS_STRIDE + k * 4 + koff];
            bfrag.x = p[0];
            bfrag.y = p[1];
            acc = __builtin_amdgcn_wmma_f32_16x16x4_f32(
                /*neg_a=*/false, afrag[k],
                /*neg_b=*/false, bfrag,
                /*c_mod=*/(short)0, acc,
                /*reuse_a=*/false, /*reuse_b=*/false);
        }
        // D layout: VGPR v -> M = v + mbase (lanes>=16: +8), N = lane&15
        #pragma unroll
        for (int v = 0; v < 8; ++v) {
            const int row = tokbase + wave * 16 + mbase + v;
            out[row * EMB + colbase + nt * 16 + m] = acc[v];
        }
    }
}

extern "C" void kernel_launch(void* const* d_in, const int* in_sizes, int n_in,
                              void* d_out, int out_size, void* d_ws, size_t ws_size,
                              hipStream_t stream) {
    const float* U0 = (const float*)d_in[0];
    const float* U1 = (const float*)d_in[1];
    const float* U2 = (const float*)d_in[2];
    const float* U3 = (const float*)d_in[3];
    const float* U4 = (const float*)d_in[4];
    const float* U5 = (const float*)d_in[5];
    const int*   x  = (const int*)  d_in[6];
    float*       out = (float*)d_out;

    const int n_tok = in_sizes[6];                   // 8192
    dim3 grid(n_tok / TOK_PER_BLOCK, EMB / COLS_PER_BLOCK);  // (64, 4)
    cp_embed_wmma_f32<<<grid, 256, 0, stream>>>(x, U0, U1, U2, U3, U4, U5, out);
}